// Critic_3693671874643
// MI455X (gfx1250) — compile-verified
//
#include <hip/hip_runtime.h>

#define B_ 1024
#define N_ 128
#define V_ 200
#define H_ 400

typedef __attribute__((ext_vector_type(2))) float v2f;
typedef __attribute__((ext_vector_type(8))) float v8f;

// Workspace layout (in floats)
#define WS_U      0
#define WS_D      (WS_U + B_ * V_)
#define WS_O      (WS_D + B_ * V_)
#define WS_SCALE  (WS_O + B_ * V_)        // 3*V folded BN scale
#define WS_SHIFT  (WS_SCALE + 3 * V_)     // 3*V folded BN shift
#define WS_HEADS  (WS_SHIFT + 3 * V_)     // B*6 heads

// ---------------- block reduction helpers (128 threads) ----------------
__device__ __forceinline__ float blockMax128(float* red, int tid, float v) {
  red[tid] = v;
  __syncthreads();
#pragma unroll
  for (int off = 64; off > 0; off >>= 1) {
    if (tid < off) red[tid] = fmaxf(red[tid], red[tid + off]);
    __syncthreads();
  }
  float r = red[0];
  __syncthreads();
  return r;
}

__device__ __forceinline__ float blockSum128(float* red, int tid, float v) {
  red[tid] = v;
  __syncthreads();
#pragma unroll
  for (int off = 64; off > 0; off >>= 1) {
    if (tid < off) red[tid] += red[tid + off];
    __syncthreads();
  }
  float r = red[0];
  __syncthreads();
  return r;
}

// ---------------------------------------------------------------------------
// Kernel 1: folded attention.  One block per batch element, 128 threads
// (one per token).  Uses  s[n] = (m[n]·(Wk@q) + q·bk)*scale  and
// out = (Σ w[n] m[n]) @ Wv + bv, avoiding the B×N×V k/v tensors entirely.
// ---------------------------------------------------------------------------
__global__ void __launch_bounds__(128) k_attend(
    const float* __restrict__ merged,
    const float* __restrict__ Wq_aug, const float* __restrict__ bq_aug,
    const float* __restrict__ Wk_aug, const float* __restrict__ bk_aug,
    const float* __restrict__ Wv_aug, const float* __restrict__ bv_aug,
    const float* __restrict__ Wq_o,  const float* __restrict__ bq_o,
    const float* __restrict__ Wk_o,  const float* __restrict__ bk_o,
    const float* __restrict__ Wv_o,  const float* __restrict__ bv_o,
    float* __restrict__ ws)
{
  __shared__ float m14[N_ * 14];
  __shared__ int   f0s[N_];
  __shared__ float red[N_];
  __shared__ float qa[V_], qo[V_];
  __shared__ float qk_a[14], qk_o7[7];
  __shared__ float qb[2];
  __shared__ float wm_u[14], wm_d[14], wm_o[7];

  const int tid = threadIdx.x;
  const int b   = blockIdx.x;
  const float* mb = merged + (size_t)b * N_ * 15;
  const float sub_bus = mb[0];   // original merged[b,0,0]
  const float sub_loc = mb[2];   // channel 2 (unmodified)

  if (tid < 14)      { wm_u[tid] = 0.f; wm_d[tid] = 0.f; }
  else if (tid < 21) { wm_o[tid - 14] = 0.f; }

  const int n = tid;
  const float* rowp = mb + n * 15;
  float c[15];
#pragma unroll
  for (int i = 0; i < 15; i++) c[i] = rowp[i];
  c[0] -= sub_bus;
  c[7] -= sub_bus;
#pragma unroll
  for (int i = 0; i < 14; i++) m14[n * 14 + i] = c[i];
  const float flag = c[14];
  const float loc  = c[2];
  const bool um = (loc < sub_loc) && (flag == 1.0f);
  const bool dm = (loc > sub_loc) && (flag == 1.0f);
  const int  f0 = (flag == 0.0f) ? 1 : 0;
  f0s[n] = f0;
  __syncthreads();

  // inclusive cumsum of f0 flags (N=128, cheap serial scan per thread)
  int cs = 0;
  for (int j = 0; j <= n; j++) cs += f0s[j];
  const bool om = (f0 != 0) && (cs > 1);

  // q vectors (ego7 = adjusted merged[b,0,0:7])
  float ego7[7];
#pragma unroll
  for (int f = 0; f < 7; f++) ego7[f] = m14[f];
  for (int d = tid; d < V_; d += 128) {
    float sa = bq_aug[d], so = bq_o[d];
#pragma unroll
    for (int f = 0; f < 7; f++) {
      sa += ego7[f] * Wq_aug[f * V_ + d];
      so += ego7[f] * Wq_o[f * V_ + d];
    }
    qa[d] = sa;
    qo[d] = so;
  }
  __syncthreads();

  // fold q through Wk:  qk[f] = Σ_d Wk[f,d] q[d] ;  qb = q·bk
  if (tid < 14) {
    float s = 0.f;
    for (int d = 0; d < V_; d++) s += Wk_aug[tid * V_ + d] * qa[d];
    qk_a[tid] = s;
  } else if (tid == 14) {
    float s = 0.f;
    for (int d = 0; d < V_; d++) s += qa[d] * bk_aug[d];
    qb[0] = s;
  } else if (tid >= 16 && tid < 23) {
    const int f = tid - 16;
    float s = 0.f;
    for (int d = 0; d < V_; d++) s += Wk_o[f * V_ + d] * qo[d];
    qk_o7[f] = s;
  } else if (tid == 23) {
    float s = 0.f;
    for (int d = 0; d < V_; d++) s += qo[d] * bk_o[d];
    qb[1] = s;
  }
  __syncthreads();

  const float scale = 0.070710678118654752f;  // 1/sqrt(200)
  float s_aug = qb[0], s_o = qb[1];
#pragma unroll
  for (int f = 0; f < 14; f++) s_aug += m14[n * 14 + f] * qk_a[f];
#pragma unroll
  for (int f = 0; f < 7; f++)  s_o   += m14[n * 14 + f] * qk_o7[f];
  s_aug *= scale;
  s_o   *= scale;

  // softmax per mask (uniform across block)
  float mv   = blockMax128(red, tid, um ? s_aug : -1e30f);
  bool  anyU = mv > -1e29f;
  float e    = (um && anyU) ? expf(s_aug - mv) : 0.f;
  float Z    = blockSum128(red, tid, e);
  float wu   = (anyU && Z > 0.f) ? e / Z : 0.f;

  mv         = blockMax128(red, tid, dm ? s_aug : -1e30f);
  bool anyD  = mv > -1e29f;
  e          = (dm && anyD) ? expf(s_aug - mv) : 0.f;
  Z          = blockSum128(red, tid, e);
  float wd   = (anyD && Z > 0.f) ? e / Z : 0.f;

  mv         = blockMax128(red, tid, om ? s_o : -1e30f);
  bool anyO  = mv > -1e29f;
  e          = (om && anyO) ? expf(s_o - mv) : 0.f;
  Z          = blockSum128(red, tid, e);
  float wo   = (anyO && Z > 0.f) ? e / Z : 0.f;

  // weighted token means  wm[f] = Σ_n w[n] m14[n,f]  (wave shuffle + LDS atomic)
  const int lane = tid & 31;
  for (int f = 0; f < 14; f++) {
    float vu = wu * m14[n * 14 + f];
    float vd = wd * m14[n * 14 + f];
#pragma unroll
    for (int off = 16; off > 0; off >>= 1) {
      vu += __shfl_xor(vu, off, 32);
      vd += __shfl_xor(vd, off, 32);
    }
    if (lane == 0) { atomicAdd(&wm_u[f], vu); atomicAdd(&wm_d[f], vd); }
  }
  for (int f = 0; f < 7; f++) {
    float vo = wo * m14[n * 14 + f];
#pragma unroll
    for (int off = 16; off > 0; off >>= 1) vo += __shfl_xor(vo, off, 32);
    if (lane == 0) atomicAdd(&wm_o[f], vo);
  }
  __syncthreads();

  // outputs: out[d] = wm @ Wv[:,d] + bv[d]  (Σw == 1), zeroed if mask empty
  float* up = ws + WS_U + (size_t)b * V_;
  float* dp = ws + WS_D + (size_t)b * V_;
  float* op = ws + WS_O + (size_t)b * V_;
  for (int d = tid; d < V_; d += 128) {
    float su = bv_aug[d], sd = bv_aug[d], so = bv_o[d];
#pragma unroll
    for (int f = 0; f < 14; f++) {
      const float w = Wv_aug[f * V_ + d];
      su += wm_u[f] * w;
      sd += wm_d[f] * w;
    }
#pragma unroll
    for (int f = 0; f < 7; f++) so += wm_o[f] * Wv_o[f * V_ + d];
    up[d] = anyU ? su : 0.f;
    dp[d] = anyD ? sd : 0.f;
    op[d] = anyO ? so : 0.f;
  }
}

// ---------------------------------------------------------------------------
// Kernel 2: batch-norm statistics, folded to per-feature affine:
//   scale = g * rsqrt(var+eps), shift = b - mean*scale.  600 features.
// ---------------------------------------------------------------------------
__global__ void __launch_bounds__(128) k_stats(
    const float* __restrict__ bn_g, const float* __restrict__ bn_b,
    float* __restrict__ ws)
{
  __shared__ float rs[128], rss[128];
  const int f   = blockIdx.x;   // 0..599
  const int t   = f / V_;
  const int j   = f % V_;
  const int tid = threadIdx.x;
  const float* src = ws + (size_t)t * B_ * V_ + j;
  float s = 0.f, ss = 0.f;
  for (int b = tid; b < B_; b += 128) {
    const float v = src[(size_t)b * V_];
    s += v;
    ss += v * v;
  }
  rs[tid] = s;
  rss[tid] = ss;
  __syncthreads();
#pragma unroll
  for (int off = 64; off > 0; off >>= 1) {
    if (tid < off) { rs[tid] += rs[tid + off]; rss[tid] += rss[tid + off]; }
    __syncthreads();
  }
  if (tid == 0) {
    const float mean = rs[0] * (1.f / B_);
    const float var  = rss[0] * (1.f / B_) - mean * mean;
    const float rstd = rsqrtf(var + 1e-5f);
    const float sc   = bn_g[j] * rstd;
    ws[WS_SCALE + f] = sc;
    ws[WS_SHIFT + f] = bn_b[j] - mean * sc;
  }
}

// ---------------------------------------------------------------------------
// Kernel 3: heads[b,k] = Σ_g elu(bn(t)@W3[k] + b3[k]) * W4[k] + b4[k]
// Grid (B/16, 6).  Block = 416 threads = 13 waves: wave w owns n-tile w
// (uniform, branch-free).  W3[k] is staged into LDS zero-padded to 208 cols
// and pre-swizzled into WMMA fragment order, so the K-loop is pure
// ds_load_b64 + v_wmma_f32_16x16x4_f32 with no predication and no vmem waits.
//   frag index fi = ks*2 + kh  (K = ks*4 + kh*2 + v)
//   B: ldsB[(fi*208 + col)*2 + v]   A: ldsA[(fi*16 + m)*2 + v]
// The two 16-lane halves read LDS regions offset by 208*8B / 16*8B -> bank
// halves 0-31 vs 32-63: conflict-free.
// ---------------------------------------------------------------------------
#define KH_THREADS 416
#define KH_BPAD    208
#define KH_LDSB    (50 * 2 * KH_BPAD * 2)   // 41600 floats
#define KH_LDSA    (50 * 2 * 16 * 2)        // 3200 floats
#define KH_LDS_BYTES ((KH_LDSB + KH_LDSA + 16) * 4)

__global__ void __launch_bounds__(KH_THREADS) k_heads(
    const float* __restrict__ W3, const float* __restrict__ b3,
    const float* __restrict__ W4, const float* __restrict__ b4,
    float* __restrict__ ws)
{
  extern __shared__ float smem[];
  float* ldsB   = smem;
  float* ldsA   = smem + KH_LDSB;
  float* rowsum = smem + KH_LDSB + KH_LDSA;

  const int tid = threadIdx.x;
  const int b0  = blockIdx.x * 16;
  const int k   = blockIdx.y;
  const int t   = k >> 1;                       // th index map [0,0,1,1,2,2]
  const float* tsrc = ws + (size_t)t * B_ * V_;
  const float* scl  = ws + WS_SCALE + t * V_;
  const float* shf  = ws + WS_SHIFT + t * V_;
  const float* W3k  = W3 + (size_t)k * V_ * V_;

  if (tid < 16) rowsum[tid] = 0.f;

  // stage B: zero-pad cols 200..207, swizzle to fragment order (coalesced reads)
  for (int idx = tid; idx < V_ * KH_BPAD; idx += KH_THREADS) {
    const int K = idx / KH_BPAD, colp = idx % KH_BPAD;
    const int colc = colp < V_ ? colp : V_ - 1;             // clamp addr, no branch
    const float v = W3k[(size_t)K * V_ + colc] * (colp < V_ ? 1.f : 0.f);
    const int ks = K >> 2, kh2 = (K >> 1) & 1, vv = K & 1;
    ldsB[(((ks * 2 + kh2) * KH_BPAD) + colp) * 2 + vv] = v;
  }
  // stage A with BN affine folded, fragment order
  for (int idx = tid; idx < 16 * V_; idx += KH_THREADS) {
    const int mm = idx / V_, K = idx % V_;
    const float v = tsrc[(size_t)(b0 + mm) * V_ + K] * scl[K] + shf[K];
    const int ks = K >> 2, kh2 = (K >> 1) & 1, vv = K & 1;
    ldsA[(((ks * 2 + kh2) * 16) + mm) * 2 + vv] = v;
  }
  __syncthreads();

  const int lane = tid & 31;
  const int wv   = tid >> 5;         // n-tile 0..12 (uniform per wave)
  const int m    = lane & 15;
  const int kh   = lane >> 4;
  const int col  = wv * 16 + m;

  v8f acc = {};
  for (int ks = 0; ks < 50; ks++) {
    const int fi = ks * 2 + kh;
    const v2f a  = *(const v2f*)&ldsA[(fi * 16 + m) * 2];
    const v2f bb = *(const v2f*)&ldsB[(fi * KH_BPAD + col) * 2];
    acc = __builtin_amdgcn_wmma_f32_16x16x4_f32(
        false, a, false, bb, (short)0, acc, false, false);
  }

  // elu + ·W4, reduce columns -> per-row head values
  const bool inb = col < V_;
  const int colc = inb ? col : V_ - 1;
  const float w4v = W4[k * V_ + colc] * (inb ? 1.f : 0.f);
  const float b3v = b3[k * V_ + colc];
#pragma unroll
  for (int r = 0; r < 8; r++) {
    const float cc = acc[r] + b3v;
    const float h  = (cc > 0.f) ? cc : (expf(cc) - 1.f);
    float contrib  = h * w4v;
#pragma unroll
    for (int off = 8; off > 0; off >>= 1)
      contrib += __shfl_xor(contrib, off, 32);   // stays within 16-lane half
    if ((lane & 15) == 0) atomicAdd(&rowsum[r + 8 * kh], contrib);
  }
  __syncthreads();
  if (tid < 16)
    ws[WS_HEADS + (size_t)(b0 + tid) * 6 + k] = rowsum[tid] + b4[k];
}

// ---------------------------------------------------------------------------
// Kernel 4: ego MLP + final outputs.
//   h1 = relu(ego4@W0+b0)   (K=4: single WMMA step, exact fit)
//   h2 = relu(h1@W1+b1)     (M=16, K=400, N=400 WMMA)
//   Q1 = h2@W2+b2 ; outputs Q1, A2, G1, G2, reg
// Block = 160 threads = 5 waves; wave w owns n-tiles w*5..w*5+4 (25 tiles of
// 16 = 400 cols exactly -> uniform, branch-free, no bounds checks).  The W1
// B-fragments for step ks+1 are issued before the ks WMMAs (software
// pipelining) so global-load latency overlaps the matrix math.
// ---------------------------------------------------------------------------
__global__ void __launch_bounds__(160) k_mlp(
    const float* __restrict__ x, const float* __restrict__ a_in,
    const float* __restrict__ W0, const float* __restrict__ b0g,
    const float* __restrict__ W1, const float* __restrict__ b1g,
    const float* __restrict__ W2, const float* __restrict__ b2g,
    const float* __restrict__ ws, float* __restrict__ out)
{
  __shared__ float ego[16 * 4];
  __shared__ float h1[16 * 401];    // 401 stride: conflict-free column reads
  __shared__ float rowsumQ[16];

  const int tid = threadIdx.x;
  const int b0  = blockIdx.x * 16;

  if (tid < 16) rowsumQ[tid] = 0.f;
  if (tid < 64) {
    const int mm = tid >> 2, cc = tid & 3;
    ego[mm * 4 + cc] =
        (cc < 3) ? x[(size_t)(b0 + mm) * N_ * 8 + 4 + cc] : a_in[b0 + mm];
  }
  // speculative L2 prefetch of W1 (emits global_prefetch_b8)
  __builtin_prefetch(W1 + (size_t)tid * 1000, 0, 0);
  __syncthreads();

  const int lane = tid & 31;
  const int wv   = tid >> 5;        // wave 0..4
  const int m    = lane & 15;
  const int kh   = lane >> 4;

  int cols[5];
#pragma unroll
  for (int ti = 0; ti < 5; ti++) cols[ti] = (wv * 5 + ti) * 16 + m;  // < 400

  // ---- layer 0: K=4 in a single WMMA per tile ----
  v2f a0;
  a0.x = ego[m * 4 + kh * 2];
  a0.y = ego[m * 4 + kh * 2 + 1];
#pragma unroll
  for (int ti = 0; ti < 5; ti++) {
    const int col = cols[ti];
    v2f bb;
    bb.x = W0[(size_t)(kh * 2) * H_ + col];
    bb.y = W0[(size_t)(kh * 2 + 1) * H_ + col];
    v8f cfrag = {};
    cfrag = __builtin_amdgcn_wmma_f32_16x16x4_f32(
        false, a0, false, bb, (short)0, cfrag, false, false);
    const float bias = b0g[col];
#pragma unroll
    for (int r = 0; r < 8; r++) {
      const float v = cfrag[r] + bias;
      h1[(r + 8 * kh) * 401 + col] = v > 0.f ? v : 0.f;
    }
  }
  __syncthreads();

  // ---- layer 1: 16x400x400 WMMA, software-pipelined B fragments ----
  v8f acc[5] = {};
  v2f bfr[5];
#pragma unroll
  for (int ti = 0; ti < 5; ti++) {
    bfr[ti].x = W1[(size_t)(kh * 2) * H_ + cols[ti]];
    bfr[ti].y = W1[(size_t)(kh * 2 + 1) * H_ + cols[ti]];
  }
  for (int ks = 0; ks < 100; ks++) {
    const int K0 = ks * 4 + kh * 2;
    v2f a;
    a.x = h1[m * 401 + K0];
    a.y = h1[m * 401 + K0 + 1];
    // issue next step's B loads before this step's WMMAs
    const int K0n = (ks < 99) ? (K0 + 4) : (kh * 2);   // clamp: harmless re-read
    v2f bnx[5];
#pragma unroll
    for (int ti = 0; ti < 5; ti++) {
      bnx[ti].x = W1[(size_t)K0n * H_ + cols[ti]];
      bnx[ti].y = W1[(size_t)(K0n + 1) * H_ + cols[ti]];
    }
#pragma unroll
    for (int ti = 0; ti < 5; ti++) {
      acc[ti] = __builtin_amdgcn_wmma_f32_16x16x4_f32(
          false, a, false, bfr[ti], (short)0, acc[ti], false, false);
    }
#pragma unroll
    for (int ti = 0; ti < 5; ti++) bfr[ti] = bnx[ti];
  }

  // relu + ·W2, reduce columns -> Q1 per row
#pragma unroll
  for (int ti = 0; ti < 5; ti++) {
    const int col = cols[ti];
    const float b1v = b1g[col];
    const float w2v = W2[col];
#pragma unroll
    for (int r = 0; r < 8; r++) {
      float v = acc[ti][r] + b1v;
      v = v > 0.f ? v : 0.f;
      float contrib = v * w2v;
#pragma unroll
      for (int off = 8; off > 0; off >>= 1)
        contrib += __shfl_xor(contrib, off, 32);
      if ((lane & 15) == 0) atomicAdd(&rowsumQ[r + 8 * kh], contrib);
    }
  }
  __syncthreads();

  if (tid < 16) {
    const int b = b0 + tid;
    const float q = rowsumQ[tid] + b2g[0];
    const float* hd = ws + WS_HEADS + (size_t)b * 6;
    const float A1 = hd[0] + hd[2] + hd[4];
    const float A2 = hd[1] + hd[3] + hd[5];
    out[b]            = q;        // Q1
    out[B_ + b]       = A2;       // A2
    out[2 * B_ + b]   = q + A1;   // G1
    out[3 * B_ + b]   = q + A2;   // G2
  }
  if (blockIdx.x == 0 && tid == 0) out[4 * B_] = 0.f;  // reg
}

// ---------------------------------------------------------------------------
extern "C" void kernel_launch(void* const* d_in, const int* in_sizes, int n_in,
                              void* d_out, int out_size, void* d_ws, size_t ws_size,
                              hipStream_t stream) {
  (void)in_sizes; (void)n_in; (void)out_size; (void)ws_size;
  const float* merged = (const float*)d_in[0];
  const float* x      = (const float*)d_in[1];
  const float* a      = (const float*)d_in[2];
  const float* Wq_aug = (const float*)d_in[3];
  const float* bq_aug = (const float*)d_in[4];
  const float* Wk_aug = (const float*)d_in[5];
  const float* bk_aug = (const float*)d_in[6];
  const float* Wv_aug = (const float*)d_in[7];
  const float* bv_aug = (const float*)d_in[8];
  const float* Wq_o   = (const float*)d_in[9];
  const float* bq_o   = (const float*)d_in[10];
  const float* Wk_o   = (const float*)d_in[11];
  const float* bk_o   = (const float*)d_in[12];
  const float* Wv_o   = (const float*)d_in[13];
  const float* bv_o   = (const float*)d_in[14];
  const float* bn_g   = (const float*)d_in[15];
  const float* bn_b   = (const float*)d_in[16];
  const float* W3     = (const float*)d_in[17];
  const float* b3     = (const float*)d_in[18];
  const float* W4     = (const float*)d_in[19];
  const float* b4     = (const float*)d_in[20];
  const float* W0     = (const float*)d_in[21];
  const float* b0     = (const float*)d_in[22];
  const float* W1     = (const float*)d_in[23];
  const float* b1     = (const float*)d_in[24];
  const float* W2     = (const float*)d_in[25];
  const float* b2     = (const float*)d_in[26];

  float* ws  = (float*)d_ws;
  float* out = (float*)d_out;

  k_attend<<<B_, 128, 0, stream>>>(merged, Wq_aug, bq_aug, Wk_aug, bk_aug,
                                   Wv_aug, bv_aug, Wq_o, bq_o, Wk_o, bk_o,
                                   Wv_o, bv_o, ws);
  k_stats<<<3 * V_, 128, 0, stream>>>(bn_g, bn_b, ws);
  k_heads<<<dim3(B_ / 16, 6), KH_THREADS, KH_LDS_BYTES, stream>>>(W3, b3, W4,
                                                                  b4, ws);
  k_mlp<<<B_ / 16, 160, 0, stream>>>(x, a, W0, b0, W1, b1, W2, b2, ws, out);
}